// LabelPropagationLoss_19327352832823
// MI455X (gfx1250) — compile-verified
//
#include <hip/hip_runtime.h>
#include <hip/hip_bf16.h>
#include <stdint.h>

#define BLOCK 256
#define TILE 256                  // edges per wave per pipeline stage
#define WPB 8                     // waves per block (256 / wave32)
#define BPB (TILE * 12)           // 3072 B per buffer: rows[1024]|cols[1024]|vals[1024]
#define ALPHA 0.5f
#define K_ITERS 10
#define POS_WEIGHT 1.0f

// ---------------- CDNA5 async global->LDS helpers (gfx1250) ----------------
__device__ __forceinline__ void async_b128(unsigned lds_off, const void* g) {
  // VDST = per-lane LDS byte address, VADDR = 64-bit global address.
  // NT hint: edge streams (204MB re-read 10x) must not evict L2-resident E.
  asm volatile("global_load_async_to_lds_b128 %0, %1, off th:TH_LOAD_NT"
               :: "v"(lds_off), "v"(g) : "memory");
}
__device__ __forceinline__ void wait_async_le6() {
  asm volatile("s_wait_asynccnt 0x6" ::: "memory");
}
__device__ __forceinline__ void wait_async_0() {
  asm volatile("s_wait_asynccnt 0x0" ::: "memory");
}

// ---------------- init: zero E, neighbor, row_sums, loss accum -------------
__global__ void init_kernel(float* __restrict__ Ea, float* __restrict__ nb,
                            float* __restrict__ rs, float* __restrict__ acc, int n) {
  int i = blockIdx.x * blockDim.x + threadIdx.x;
  int n2 = 2 * n;
  if (i < n2) { Ea[i] = 0.f; nb[i] = 0.f; }
  if (i < n)  rs[i] = 0.f;
  if (i < 2)  acc[i] = 0.f;
}

// E[pos,1]=1 ; E[neg,0]=1 (distinct columns -> no conflict with each other)
__global__ void labels_kernel(float* __restrict__ Ea, const int* __restrict__ pos,
                              const int* __restrict__ neg, int p) {
  int i = blockIdx.x * blockDim.x + threadIdx.x;
  if (i < p) {
    Ea[2 * pos[i] + 1] = 1.0f;
    Ea[2 * neg[i] + 0] = 1.0f;
  }
}

// row_sums via 4-edge run-merged atomics (rows sorted in the 16M prefix)
__global__ void rowsum_kernel(const int* __restrict__ rows, const float* __restrict__ vals,
                              float* __restrict__ rs, int nnz) {
  int q = blockIdx.x * blockDim.x + threadIdx.x;
  int i0 = q * 4;
  if (i0 >= nnz) return;
  int end = min(i0 + 4, nnz);
  int cur = __builtin_nontemporal_load(&rows[i0]);
  float a = 0.f;
  for (int i = i0; i < end; ++i) {
    int r   = __builtin_nontemporal_load(&rows[i]);
    float v = __builtin_nontemporal_load(&vals[i]);
    if (r != cur) { atomicAdd(&rs[cur], a); cur = r; a = 0.f; }
    a += v;
  }
  atomicAdd(&rs[cur], a);
}

__global__ void recip_kernel(float* __restrict__ rs, int n) {
  int i = blockIdx.x * blockDim.x + threadIdx.x;
  if (i < n) rs[i] = 1.0f / fmaxf(rs[i], 1e-12f);  // in-place -> d_inv
}

// ------- SpMM scatter: async-LDS double-buffered edge streaming ------------
#define FLUSH_RUN()                                            \
  do { atomicAdd(&nb[2 * cur + 0], a0);                        \
       atomicAdd(&nb[2 * cur + 1], a1); } while (0)

#define STEP(R, C, V)                                          \
  do { int _r = (R);                                           \
       if (_r != cur) { FLUSH_RUN(); cur = _r; a0 = 0.f; a1 = 0.f; } \
       float2 _e = E2[(C)];                                    \
       a0 += (V) * _e.x; a1 += (V) * _e.y; } while (0)

__global__ void __launch_bounds__(BLOCK)
spmm_scatter_async(const int* __restrict__ rows, const int* __restrict__ cols,
                   const float* __restrict__ vals, const float2* __restrict__ E2,
                   float* __restrict__ nb, int ntiles) {
  __shared__ __align__(16) char smem[WPB * 2 * BPB];   // 48 KB / block
  const int lane   = threadIdx.x & 31;
  const int wslot  = threadIdx.x >> 5;
  const int wid    = blockIdx.x * WPB + wslot;
  const int nwaves = gridDim.x * WPB;
  char* buf = smem + (size_t)wslot * (2 * BPB);

  int t = wid;
  if (t >= ntiles) return;

  // stage tile tt into buffer b: 6 async b128 loads per lane
  auto issue = [&](int tt, int b) {
    const int e0 = tt * TILE + lane * 4;   // first 128-edge half, 4 elems/lane
    const unsigned lb =
        (unsigned)(uintptr_t)(buf + b * BPB) + (unsigned)(lane * 16);
    async_b128(lb,          rows + e0);
    async_b128(lb +  512,   rows + e0 + 128);
    async_b128(lb + 1024,   cols + e0);
    async_b128(lb + 1536,   cols + e0 + 128);
    async_b128(lb + 2048,   vals + e0);
    async_b128(lb + 2560,   vals + e0 + 128);
  };

  int pb = 0;
  issue(t, 0);
  while (t < ntiles) {
    const int tn = t + nwaves;
    if (tn < ntiles) { issue(tn, pb ^ 1); wait_async_le6(); }  // next 6 stay in flight
    else             { wait_async_0(); }

    // lane processes 8 consecutive edges [t*TILE + 8*lane, +7]
    const char* p = buf + pb * BPB;
    const int4   r0 = ((const int4*)(p        ))[2 * lane];
    const int4   r1 = ((const int4*)(p        ))[2 * lane + 1];
    const int4   c0 = ((const int4*)(p + 1024))[2 * lane];
    const int4   c1 = ((const int4*)(p + 1024))[2 * lane + 1];
    const float4 v0 = ((const float4*)(p + 2048))[2 * lane];
    const float4 v1 = ((const float4*)(p + 2048))[2 * lane + 1];

    int cur = r0.x;
    float a0 = 0.f, a1 = 0.f;
    STEP(r0.x, c0.x, v0.x);
    STEP(r0.y, c0.y, v0.y);
    STEP(r0.z, c0.z, v0.z);
    STEP(r0.w, c0.w, v0.w);
    STEP(r1.x, c1.x, v1.x);
    STEP(r1.y, c1.y, v1.y);
    STEP(r1.z, c1.z, v1.z);
    STEP(r1.w, c1.w, v1.w);
    FLUSH_RUN();

    t = tn;
    pb ^= 1;
  }
}

// tail edges (NNZ % TILE) via the simple path
__global__ void spmm_scatter_tail(const int* __restrict__ rows, const int* __restrict__ cols,
                                  const float* __restrict__ vals, const float2* __restrict__ E2,
                                  float* __restrict__ nb, int start, int nnz) {
  int i = start + blockIdx.x * blockDim.x + threadIdx.x;
  if (i < nnz) {
    int   c = __builtin_nontemporal_load(&cols[i]);
    int   r = __builtin_nontemporal_load(&rows[i]);
    float v = __builtin_nontemporal_load(&vals[i]);
    float2 e = E2[c];
    atomicAdd(&nb[2 * r + 0], v * e.x);
    atomicAdd(&nb[2 * r + 1], v * e.y);
  }
}

// fused: E_new = softmax((alpha*E + (1-alpha)*d_inv*nb)/T); reset nb for next iter
__global__ void update_kernel(const float2* __restrict__ Esrc, float2* __restrict__ Edst,
                              float2* __restrict__ nb2, const float* __restrict__ dinv,
                              const float* __restrict__ temp, int n) {
  int i = blockIdx.x * blockDim.x + threadIdx.x;
  if (i >= n) return;
  float2 e = Esrc[i];
  float2 b = nb2[i];
  nb2[i] = make_float2(0.f, 0.f);            // zero neighbor for next iteration
  float di = dinv[i];
  float it = 1.0f / temp[0];
  float z0 = (ALPHA * e.x + (1.0f - ALPHA) * di * b.x) * it;
  float z1 = (ALPHA * e.y + (1.0f - ALPHA) * di * b.y) * it;
  float m  = fmaxf(z0, z1);
  float p0 = expf(z0 - m);
  float p1 = expf(z1 - m);
  float s  = 1.0f / (p0 + p1);
  Edst[i] = make_float2(p0 * s, p1 * s);
}

// loss: block-reduced partial sums of -log(clip(prob)), atomic combine
__global__ void loss_kernel(const float* __restrict__ E, const int* __restrict__ pos,
                            const int* __restrict__ neg, float* __restrict__ acc, int p) {
  __shared__ float s0[BLOCK], s1[BLOCK];
  float lp = 0.f, ln = 0.f;
  for (int i = blockIdx.x * blockDim.x + threadIdx.x; i < p;
       i += gridDim.x * blockDim.x) {
    lp -= logf(fmaxf(E[2 * pos[i] + 1], 1e-6f));
    ln -= logf(fmaxf(E[2 * neg[i] + 0], 1e-6f));
  }
  s0[threadIdx.x] = lp; s1[threadIdx.x] = ln;
  __syncthreads();
  for (int o = BLOCK / 2; o > 0; o >>= 1) {
    if (threadIdx.x < o) {
      s0[threadIdx.x] += s0[threadIdx.x + o];
      s1[threadIdx.x] += s1[threadIdx.x + o];
    }
    __syncthreads();
  }
  if (threadIdx.x == 0) { atomicAdd(&acc[0], s0[0]); atomicAdd(&acc[1], s1[0]); }
}

__global__ void finalize_kernel(const float* __restrict__ acc, float* __restrict__ out, int p) {
  if (blockIdx.x == 0 && threadIdx.x == 0) {
    float invP = 1.0f / (float)p;
    out[0] = POS_WEIGHT * (acc[0] * invP) + (acc[1] * invP);
  }
}

__global__ void copyE_kernel(const float* __restrict__ Efin, float* __restrict__ out, int n2) {
  int i = blockIdx.x * blockDim.x + threadIdx.x;
  if (i < n2) __builtin_nontemporal_store(Efin[i], &out[i]);
}

// ---------------------------------------------------------------------------
extern "C" void kernel_launch(void* const* d_in, const int* in_sizes, int n_in,
                              void* d_out, int out_size, void* d_ws, size_t ws_size,
                              hipStream_t stream) {
  (void)n_in; (void)out_size;
  // inputs: embeddings, positive_nodes, negative_nodes, rows, cols, vals, temperature
  const int*   pos  = (const int*)d_in[1];
  const int*   neg  = (const int*)d_in[2];
  const int*   rows = (const int*)d_in[3];
  const int*   cols = (const int*)d_in[4];
  const float* vals = (const float*)d_in[5];
  const float* temp = (const float*)d_in[6];
  const int N   = in_sizes[0] / 64;   // embeddings is [N,64]; only N is used
  const int P   = in_sizes[1];
  const int NNZ = in_sizes[3];
  float* out = (float*)d_out;

  // workspace layout (floats): rs/d_inv[N] | Ea[2N] | Eb[2N] | nb[2N] | acc[2]
  if (ws_size < ((size_t)7 * (size_t)N + 2) * sizeof(float)) return;
  float* rs  = (float*)d_ws;
  float* Ea  = rs + N;
  float* Eb  = Ea + 2 * (size_t)N;
  float* nb  = Eb + 2 * (size_t)N;
  float* acc = nb + 2 * (size_t)N;

  const int n2 = 2 * N;
  init_kernel<<<(n2 + BLOCK - 1) / BLOCK, BLOCK, 0, stream>>>(Ea, nb, rs, acc, N);
  labels_kernel<<<(P + BLOCK - 1) / BLOCK, BLOCK, 0, stream>>>(Ea, pos, neg, P);

  const int quads = (NNZ + 3) / 4;
  rowsum_kernel<<<(quads + BLOCK - 1) / BLOCK, BLOCK, 0, stream>>>(rows, vals, rs, NNZ);
  recip_kernel<<<(N + BLOCK - 1) / BLOCK, BLOCK, 0, stream>>>(rs, N);

  const int ntiles = NNZ / TILE;
  const int tail_start = ntiles * TILE;
  float* src = Ea;
  float* dst = Eb;
  for (int k = 0; k < K_ITERS; ++k) {
    spmm_scatter_async<<<1024, BLOCK, 0, stream>>>(rows, cols, vals,
                                                   (const float2*)src, nb, ntiles);
    if (tail_start < NNZ) {
      spmm_scatter_tail<<<(NNZ - tail_start + BLOCK - 1) / BLOCK, BLOCK, 0, stream>>>(
          rows, cols, vals, (const float2*)src, nb, tail_start, NNZ);
    }
    update_kernel<<<(N + BLOCK - 1) / BLOCK, BLOCK, 0, stream>>>(
        (const float2*)src, (float2*)dst, (float2*)nb, rs, temp, N);
    float* tswap = src; src = dst; dst = tswap;
  }

  loss_kernel<<<256, BLOCK, 0, stream>>>(src, pos, neg, acc, P);
  finalize_kernel<<<1, 32, 0, stream>>>(acc, out, P);
  copyE_kernel<<<(n2 + BLOCK - 1) / BLOCK, BLOCK, 0, stream>>>(src, out + 1, n2);
}